// PiecewiseLinearUnitV2_43215960933189
// MI455X (gfx1250) — compile-verified
//
#include <hip/hip_runtime.h>

#define MAX_N 20

typedef float v4f __attribute__((ext_vector_type(4)));
typedef float v2f __attribute__((ext_vector_type(2)));

// Per-element piecewise-linear evaluation. All divisions hoisted out; one
// conflict-free LDS b64 gather ({Y0, slope}) + ~13 VALU ops per element.
__device__ __forceinline__ float plu_eval(float xx,
                                          float Bl, float Br,
                                          float Kl, float Kr,
                                          float len, float inv_len,
                                          float nh0, float top,
                                          const v2f* sh_YK) {
    float tL    = xx - Bl;                       // x - Bl
    float idx_f = floorf(tL * inv_len);          // bin index (float)
    int   idx   = (int)idx_f;
    idx = idx < 0 ? 0 : (idx > (MAX_N - 1) ? (MAX_N - 1) : idx);  // v_med3_i32
    v2f yk = sh_YK[idx];                         // ds_load_b64: {Y0, Kidx}
    // l3 = (x - Bidx)*Kidx + Y0, with x - Bidx = tL - idx_f*len
    float l3 = fmaf(fmaf(-idx_f, len, tL), yk.y, yk.x);
    float l1 = fmaf(tL, Kl, nh0);                // left linear extension
    float l2 = fmaf(xx - Br, Kr, top);           // right linear extension
    return (xx < Bl) ? l1 : ((xx >= Br) ? l2 : l3);
}

__device__ __forceinline__ v4f plu_eval4(v4f xv,
                                         float Bl, float Br,
                                         float Kl, float Kr,
                                         float len, float inv_len,
                                         float nh0, float top,
                                         const v2f* sh_YK) {
    v4f rv;
#pragma unroll
    for (int j = 0; j < 4; ++j) {
        rv[j] = plu_eval(xv[j], Bl, Br, Kl, Kr, len, inv_len, nh0, top, sh_YK);
    }
    return rv;
}

// 256 threads/block (8 wave32s), 8 elements/thread = 2048 elements/block.
__global__ __launch_bounds__(256)
void PiecewiseLinearUnitV2_kernel(const float* __restrict__ x,
                                  const float* __restrict__ Np,
                                  const float* __restrict__ Bounds,
                                  const float* __restrict__ BoundSlope,
                                  const float* __restrict__ nheight,
                                  float* __restrict__ out,
                                  int n4, int tail, long long ntotal) {
    // LUT: 20 float2 entries -> 40 distinct LDS banks -> conflict-free gather.
    __shared__ v2f   sh_YK[MAX_N];
    __shared__ float sh_par[8];   // Bl, Br, Kl, Kr, len, 1/len, nheight[0], top

    const int t = threadIdx.x;
    if (t < MAX_N) {
        // Uniform-address loads: one request per wave, L2/L0 hit.
        float N         = Np[0];
        float intervals = floorf(fminf(fmaxf(N, 3.0f), (float)MAX_N));
        float Bl        = Bounds[0];
        float Br        = Bounds[1];
        float len       = (Br - Bl) / intervals;   // divide 20x total, not 51M x
        float y0 = nheight[t];
        float y1 = nheight[t + 1];
        v2f yk;
        yk.x = y0;
        yk.y = (y1 - y0) / len;                    // per-bin slope
        sh_YK[t] = yk;
        if (t == 0) {
            sh_par[0] = Bl;
            sh_par[1] = Br;
            sh_par[2] = BoundSlope[0];
            sh_par[3] = BoundSlope[1];
            sh_par[4] = len;
            sh_par[5] = 1.0f / len;
            sh_par[6] = nheight[0];
            int itop = (int)(intervals + 1.0f);
            itop = itop > MAX_N ? MAX_N : itop;    // mimic JAX clip-mode gather
            sh_par[7] = nheight[itop];
        }
    }
    __syncthreads();

    const float Bl = sh_par[0], Br = sh_par[1];
    const float Kl = sh_par[2], Kr = sh_par[3];
    const float len = sh_par[4], inv_len = sh_par[5];
    const float nh0 = sh_par[6], top = sh_par[7];

    const v4f* __restrict__ x4 = (const v4f*)x;
    v4f*       __restrict__ o4 = (v4f*)out;

    // Two float4s per thread, coalesced within the block (stride 256 vecs).
    const int i0 = blockIdx.x * 512 + t;
    const int i1 = i0 + 256;

    // Issue both 128-bit NT loads before computing: 2 outstanding LOADcnt
    // entries per wave -> better HBM latency hiding. Non-temporal so the
    // ~205MB read-once/write-once stream doesn't churn the 192MB L2.
    v4f xv0, xv1;
    const bool p0 = (i0 < n4);
    const bool p1 = (i1 < n4);
    if (p0) xv0 = __builtin_nontemporal_load(&x4[i0]);
    if (p1) xv1 = __builtin_nontemporal_load(&x4[i1]);

    if (p0) {
        v4f rv0 = plu_eval4(xv0, Bl, Br, Kl, Kr, len, inv_len, nh0, top, sh_YK);
        __builtin_nontemporal_store(rv0, &o4[i0]);
    }
    if (p1) {
        v4f rv1 = plu_eval4(xv1, Bl, Br, Kl, Kr, len, inv_len, nh0, top, sh_YK);
        __builtin_nontemporal_store(rv1, &o4[i1]);
    }

    // Scalar tail (n is 128-divisible for this shape, but stay generic).
    if (blockIdx.x == 0 && t < tail) {
        long long k = ntotal - (long long)tail + (long long)t;
        out[k] = plu_eval(x[k], Bl, Br, Kl, Kr, len, inv_len, nh0, top, sh_YK);
    }
}

extern "C" void kernel_launch(void* const* d_in, const int* in_sizes, int n_in,
                              void* d_out, int out_size, void* d_ws, size_t ws_size,
                              hipStream_t stream) {
    // setup_inputs() order: x, N, Bounds, BoundSlope, nheight  (all float32)
    const float* x          = (const float*)d_in[0];
    const float* Np         = (const float*)d_in[1];
    const float* Bounds     = (const float*)d_in[2];
    const float* BoundSlope = (const float*)d_in[3];
    const float* nheight    = (const float*)d_in[4];
    float*       out        = (float*)d_out;

    long long n    = (long long)out_size;
    int       n4   = (int)(n / 4);
    int       tail = (int)(n - 4LL * (long long)n4);
    int       blocks = (n4 + 511) / 512;   // 512 float4s = 2048 elems per block
    if (blocks < 1) blocks = 1;

    PiecewiseLinearUnitV2_kernel<<<blocks, 256, 0, stream>>>(
        x, Np, Bounds, BoundSlope, nheight, out, n4, tail, n);
}